// GraphConvolution_2_24644522344645
// MI455X (gfx1250) — compile-verified
//
#include <hip/hip_runtime.h>

#define D_IN  256
#define D_OUT 64

typedef float v2f __attribute__((ext_vector_type(2)));
typedef float v8f __attribute__((ext_vector_type(8)));

// ---------------------------------------------------------------------------
// Kernel 1: h = x @ W  via V_WMMA_F32_16X16X4_F32
// block = 128 threads (4 waves). Each wave -> 16 rows x 64 cols of h.
// W (256x64 = 64KB) staged transposed in LDS: Wt[n][k] so B-fragments are
// contiguous float2 -> ds_load_b64.
// ---------------------------------------------------------------------------
__global__ void __launch_bounds__(128)
gemm_xw_wmma(const float* __restrict__ x, const float* __restrict__ W,
             float* __restrict__ h, int n_nodes)
{
    __shared__ float Wt[D_OUT * D_IN];   // 64 KB, transposed [n][k]

    const int t = threadIdx.x;
    // Cooperative transposed load of W into LDS (one-time, tiny vs main loop)
    for (int idx = t; idx < D_IN * D_OUT; idx += 128) {
        int k = idx >> 6;     // idx / 64  (K index, row of W)
        int n = idx & 63;     // idx % 64  (N index, col of W)
        Wt[n * D_IN + k] = W[idx];
    }
    __syncthreads();

    const int lane = t & 31;
    const int wave = t >> 5;
    const int rowBase = blockIdx.x * 64 + wave * 16;

    const int m    = lane & 15;            // M (lanes 0-15) per ISA A layout
    const int koff = (lane >> 4) << 1;     // lanes 0-15 -> K+0,K+1 ; 16-31 -> K+2,K+3

    int row = rowBase + m;
    int rclamp = row < n_nodes ? row : (n_nodes - 1);  // clamp: keep EXEC all-1s
    const float* xr = x + (size_t)rclamp * D_IN + koff;

    v8f c0 = {}; v8f c1 = {}; v8f c2 = {}; v8f c3 = {};

    const float* wp = &Wt[m * D_IN + koff];

#pragma unroll 4
    for (int k = 0; k < D_IN; k += 4) {
        v2f a  = *(const v2f*)(xr + k);                       // A frag (global)
        v2f b0 = *(const v2f*)(wp + k + 0 * 16 * D_IN);       // B frags (LDS)
        v2f b1 = *(const v2f*)(wp + k + 1 * 16 * D_IN);
        v2f b2 = *(const v2f*)(wp + k + 2 * 16 * D_IN);
        v2f b3 = *(const v2f*)(wp + k + 3 * 16 * D_IN);
        c0 = __builtin_amdgcn_wmma_f32_16x16x4_f32(false, a, false, b0, (short)0, c0, false, false);
        c1 = __builtin_amdgcn_wmma_f32_16x16x4_f32(false, a, false, b1, (short)0, c1, false, false);
        c2 = __builtin_amdgcn_wmma_f32_16x16x4_f32(false, a, false, b2, (short)0, c2, false, false);
        c3 = __builtin_amdgcn_wmma_f32_16x16x4_f32(false, a, false, b3, (short)0, c3, false, false);
    }

    // C/D layout: VGPR v = row M=v (lanes 0-15) or M=v+8 (lanes 16-31); N = lane&15
    const int mrow = rowBase + ((lane >> 4) << 3);
    const int ncol = lane & 15;
#pragma unroll
    for (int v = 0; v < 8; ++v) {
        int r = mrow + v;
        if (r < n_nodes) {
            float* hp = h + (size_t)r * D_OUT + ncol;
            hp[0]  = c0[v];
            hp[16] = c1[v];
            hp[32] = c2[v];
            hp[48] = c3[v];
        }
    }
}

// ---------------------------------------------------------------------------
// Kernel 2: edge scatter-gather. One wave (32 lanes) per edge; each lane owns
// 2 of the 64 features: float2 gather of h[col] (L2-resident: h = 25.6MB
// << 192MB L2) and two hardware global_atomic_add_f32 into out[row].
// ---------------------------------------------------------------------------
__global__ void __launch_bounds__(256)
spmm_edge_kernel(const float* __restrict__ h, const int* __restrict__ erow,
                 const int* __restrict__ ecol, const float* __restrict__ eval,
                 float* __restrict__ out, int n_edges)
{
    int gwave = (int)((blockIdx.x * 256u + threadIdx.x) >> 5);
    int lane  = threadIdx.x & 31;
    if (gwave >= n_edges) return;

    int   c = ecol[gwave];
    int   r = erow[gwave];
    float v = eval[gwave];

    v2f hv = *(const v2f*)(h + (size_t)c * D_OUT + lane * 2);
    float* op = out + (size_t)r * D_OUT + lane * 2;
    unsafeAtomicAdd(op + 0, hv.x * v);   // global_atomic_add_f32 (no return)
    unsafeAtomicAdd(op + 1, hv.y * v);
}

// ---------------------------------------------------------------------------
// Kernel 3: in-place ReLU on out
// ---------------------------------------------------------------------------
__global__ void __launch_bounds__(256)
relu_kernel(float* __restrict__ out, int n)
{
    int i = blockIdx.x * 256 + threadIdx.x;
    if (i < n) out[i] = fmaxf(out[i], 0.0f);
}

// ---------------------------------------------------------------------------
extern "C" void kernel_launch(void* const* d_in, const int* in_sizes, int n_in,
                              void* d_out, int out_size, void* d_ws, size_t ws_size,
                              hipStream_t stream)
{
    const float* x    = (const float*)d_in[0];
    const float* W    = (const float*)d_in[1];
    const int*   erow = (const int*)d_in[2];
    const int*   ecol = (const int*)d_in[3];
    const float* eval = (const float*)d_in[4];
    float* out = (float*)d_out;
    float* h   = (float*)d_ws;               // 100000*64*4 = 25.6 MB scratch

    const int n_nodes = in_sizes[0] / D_IN;
    const int n_edges = in_sizes[2];
    const int out_elems = n_nodes * D_OUT;

    // 1) h = x @ W  (WMMA)
    int gemm_blocks = (n_nodes + 63) / 64;
    gemm_xw_wmma<<<gemm_blocks, 128, 0, stream>>>(x, W, h, n_nodes);

    // 2) zero the accumulator (capture-safe async memset node)
    hipMemsetAsync(d_out, 0, (size_t)out_elems * sizeof(float), stream);

    // 3) edge gather + atomic scatter-add (1 wave per edge -> 8 edges/block)
    int edge_blocks = (n_edges + 7) / 8;
    spmm_edge_kernel<<<edge_blocks, 256, 0, stream>>>(h, erow, ecol, eval, out, n_edges);

    // 4) ReLU in place
    int relu_blocks = (out_elems + 255) / 256;
    relu_kernel<<<relu_blocks, 256, 0, stream>>>(out, out_elems);
}